// NSGCN_30691836297929
// MI455X (gfx1250) — compile-verified
//
#include <hip/hip_runtime.h>

// ---------------------------------------------------------------------------
// 3-layer GCN (DGL GraphConv norm='both') for MI455X / gfx1250.
// Dense GEMMs use V_WMMA_F32_16X16X4_F32 (native f32 WMMA, keeps reference
// precision). SpMM is edge-parallel atomicAdd into L2-resident feature maps.
// ---------------------------------------------------------------------------

typedef float v2f __attribute__((ext_vector_type(2)));
typedef float v8f __attribute__((ext_vector_type(8)));

#define N_FEAT_IN 100

// ---------------- degree accumulation ----------------
__global__ void degree_kernel(const int* __restrict__ src,
                              const int* __restrict__ dst,
                              float* __restrict__ deg_out,
                              float* __restrict__ deg_in,
                              int n_edges) {
    int e = blockIdx.x * blockDim.x + threadIdx.x;
    if (e >= n_edges) return;
    atomicAdd(&deg_out[src[e]], 1.0f);
    atomicAdd(&deg_in[dst[e]], 1.0f);
}

// in-place: deg -> rsqrt(max(deg, 1))
__global__ void norm_kernel(float* __restrict__ a, float* __restrict__ b, int n) {
    int i = blockIdx.x * blockDim.x + threadIdx.x;
    if (i >= n) return;
    a[i] = rsqrtf(fmaxf(a[i], 1.0f));
    b[i] = rsqrtf(fmaxf(b[i], 1.0f));
}

// ---------------- SpMM scatter: agg[dst] += h[src] * nsrc[src] -------------
// One thread per (edge, 4-feature chunk). Feature rows are 16B aligned
// (F=100 -> 400B rows, F=64 -> 256B rows), so float4 gathers are legal.
template <int F>
__global__ void scatter_add_kernel(const float* __restrict__ h,
                                   const float* __restrict__ nsrc,
                                   const int* __restrict__ src,
                                   const int* __restrict__ dst,
                                   float* __restrict__ agg,
                                   int n_edges) {
    constexpr int CH = F / 4;
    long long idx = (long long)blockIdx.x * blockDim.x + threadIdx.x;
    if (idx >= (long long)n_edges * CH) return;
    int e = (int)(idx / CH);
    int f = (int)(idx % CH) * 4;
    int s = src[e];
    int d = dst[e];
    float ns = nsrc[s];
    float4 v = *(const float4*)(h + (size_t)s * F + f);
    float* arow = agg + (size_t)d * F + f;
    atomicAdd(arow + 0, v.x * ns);
    atomicAdd(arow + 1, v.y * ns);
    atomicAdd(arow + 2, v.z * ns);
    atomicAdd(arow + 3, v.w * ns);
}

// ---------------- WMMA GEMM layer ------------------------------------------
// out[r, c] = act( ndst[r] * sum_k agg[r,k] * W[k,c] + b[c] )
// One wave (32 lanes) per 16-row tile; NT column tiles of 16 held in NT v8f
// accumulators so the A fragment is reused across all V_WMMA issues per k-step.
//
// Fragment layouts (ISA 7.12.2, wave32):
//   A 16x4 f32 : lane L (m = L&15, half = L>>4): v[0]=A[m][k0+2*half],
//                                                v[1]=A[m][k0+2*half+1]
//   B 4x16 f32 : lane L (n = L&15, half = L>>4): v[0]=B[k0+2*half][n],
//                                                v[1]=B[k0+2*half+1][n]
//   C/D 16x16  : vgpr i, lane L: D[i + 8*half][L&15]
template <int F_IN, int F_OUT, int NT, bool RELU>
__global__ __launch_bounds__(32) void gcn_wmma_layer(
    const float* __restrict__ agg, const float* __restrict__ ndst,
    const float* __restrict__ W, const float* __restrict__ bias,
    float* __restrict__ out, int n_nodes) {
    const int lane = threadIdx.x;
    const int half = lane >> 4;
    const int lm = lane & 15;
    const int row0 = blockIdx.x * 16;

    v8f acc[NT];
#pragma unroll
    for (int t = 0; t < NT; ++t) acc[t] = v8f{};

    const float* arow = agg + (size_t)(row0 + lm) * F_IN;

    for (int k0 = 0; k0 < F_IN; k0 += 4) {
        const int ka = k0 + 2 * half;
        v2f a;
        a.x = arow[ka];
        a.y = arow[ka + 1];
#pragma unroll
        for (int t = 0; t < NT; ++t) {
            const int col = t * 16 + lm;
            v2f bf;
            bf.x = (col < F_OUT) ? W[(size_t)ka * F_OUT + col] : 0.0f;
            bf.y = (col < F_OUT) ? W[(size_t)(ka + 1) * F_OUT + col] : 0.0f;
            acc[t] = __builtin_amdgcn_wmma_f32_16x16x4_f32(
                /*neg_a=*/false, a, /*neg_b=*/false, bf,
                /*c_mod=*/(short)0, acc[t], /*reuse_a=*/false, /*reuse_b=*/false);
        }
    }

    // Epilogue: fold in D_in^{-1/2} row scale (commutes with GEMM), bias, ReLU.
#pragma unroll
    for (int i = 0; i < 8; ++i) {
        const int r = row0 + i + 8 * half;
        const float nd = ndst[r];
#pragma unroll
        for (int t = 0; t < NT; ++t) {
            const int col = t * 16 + lm;
            if (col < F_OUT) {
                float v = acc[t][i] * nd + bias[col];
                if (RELU) v = fmaxf(v, 0.0f);
                out[(size_t)r * F_OUT + col] = v;
            }
        }
    }
}

// ---------------------------------------------------------------------------
static inline size_t align256(size_t x) { return (x + 255) & ~(size_t)255; }

extern "C" void kernel_launch(void* const* d_in, const int* in_sizes, int n_in,
                              void* d_out, int out_size, void* d_ws, size_t ws_size,
                              hipStream_t stream) {
    const float* x  = (const float*)d_in[0];
    const float* W0 = (const float*)d_in[1];
    const float* b0 = (const float*)d_in[2];
    const float* W1 = (const float*)d_in[3];
    const float* b1 = (const float*)d_in[4];
    const float* W2 = (const float*)d_in[5];
    const float* b2 = (const float*)d_in[6];
    const int* src  = (const int*)d_in[7];
    const int* dst  = (const int*)d_in[8];

    const int N = in_sizes[0] / N_FEAT_IN;  // 50000 (multiple of 16)
    const int E = in_sizes[7];              // 800000
    float* out = (float*)d_out;

    // workspace carve-up (256B aligned)
    char* p = (char*)d_ws;
    float* nsrc = (float*)p;  p += align256((size_t)N * 4);
    float* ndst = (float*)p;  p += align256((size_t)N * 4);
    float* agg  = (float*)p;  p += align256((size_t)N * N_FEAT_IN * 4);
    float* h0   = (float*)p;  p += align256((size_t)N * 64 * 4);
    float* h1   = (float*)p;  p += align256((size_t)N * 64 * 4);

    const int TB = 256;

    // ---- degree norms -----------------------------------------------------
    hipMemsetAsync(nsrc, 0, (size_t)N * 4, stream);
    hipMemsetAsync(ndst, 0, (size_t)N * 4, stream);
    degree_kernel<<<(E + TB - 1) / TB, TB, 0, stream>>>(src, dst, nsrc, ndst, E);
    norm_kernel<<<(N + TB - 1) / TB, TB, 0, stream>>>(nsrc, ndst, N);

    // ---- layer 0: [N,100] -> [N,64], ReLU ---------------------------------
    hipMemsetAsync(agg, 0, (size_t)N * N_FEAT_IN * 4, stream);
    {
        long long total = (long long)E * (N_FEAT_IN / 4);
        int blocks = (int)((total + TB - 1) / TB);
        scatter_add_kernel<N_FEAT_IN><<<blocks, TB, 0, stream>>>(x, nsrc, src, dst, agg, E);
    }
    gcn_wmma_layer<N_FEAT_IN, 64, 4, true>
        <<<N / 16, 32, 0, stream>>>(agg, ndst, W0, b0, h0, N);

    // ---- layer 1: [N,64] -> [N,64], ReLU ----------------------------------
    hipMemsetAsync(agg, 0, (size_t)N * 64 * 4, stream);
    {
        long long total = (long long)E * (64 / 4);
        int blocks = (int)((total + TB - 1) / TB);
        scatter_add_kernel<64><<<blocks, TB, 0, stream>>>(h0, nsrc, src, dst, agg, E);
    }
    gcn_wmma_layer<64, 64, 4, true>
        <<<N / 16, 32, 0, stream>>>(agg, ndst, W1, b1, h1, N);

    // ---- layer 2: [N,64] -> [N,47], no activation -------------------------
    hipMemsetAsync(agg, 0, (size_t)N * 64 * 4, stream);
    {
        long long total = (long long)E * (64 / 4);
        int blocks = (int)((total + TB - 1) / TB);
        scatter_add_kernel<64><<<blocks, TB, 0, stream>>>(h1, nsrc, src, dst, agg, E);
    }
    gcn_wmma_layer<64, 47, 3, false>
        <<<N / 16, 32, 0, stream>>>(agg, ndst, W2, b2, out, N);
}